// LightGCN_36455682408567
// MI455X (gfx1250) — compile-verified
//
#include <hip/hip_runtime.h>

#define N_USERS_C 100000
#define N_ITEMS_C 200000
#define EMB 256
#define N_EDGES_C 1000000
#define BATCH_C 8192
#define NUM_NODES (N_USERS_C + N_ITEMS_C)

typedef __attribute__((ext_vector_type(2))) float v2f;
typedef __attribute__((ext_vector_type(8))) float v8f;

// ---------------------------------------------------------------------------
// Degree accumulation: one thread per undirected edge, two fp32 atomic adds.
// ---------------------------------------------------------------------------
__global__ void lgcn_deg_kernel(const int* __restrict__ erow,
                                const int* __restrict__ ecol,
                                float* __restrict__ deg) {
  int e = blockIdx.x * blockDim.x + threadIdx.x;
  if (e >= N_EDGES_C) return;
  unsafeAtomicAdd(&deg[erow[e]], 1.0f);
  unsafeAtomicAdd(&deg[N_USERS_C + ecol[e]], 1.0f);
}

// deg -> deg^(-1/2) in place (deg clamped to >= 1)
__global__ void lgcn_dinv_kernel(float* __restrict__ deg) {
  int n = blockIdx.x * blockDim.x + threadIdx.x;
  if (n >= NUM_NODES) return;
  deg[n] = rsqrtf(fmaxf(deg[n], 1.0f));
}

// ---------------------------------------------------------------------------
// Layer-0 gathers: ue/pe/ne accumulators = raw embedding rows; also writes the
// three raw-gather output sections. One wave per row, 2x float4 per lane.
// ---------------------------------------------------------------------------
__global__ void lgcn_init_gather(const float* __restrict__ user_emb,
                                 const float* __restrict__ item_emb,
                                 const int* __restrict__ uidx,
                                 const int* __restrict__ pidx,
                                 const int* __restrict__ nidx,
                                 float* __restrict__ ue,
                                 float* __restrict__ pe,
                                 float* __restrict__ ne,
                                 float* __restrict__ out) {
  int wid  = (blockIdx.x * blockDim.x + threadIdx.x) >> 5;  // 0..3*BATCH-1
  int lane = threadIdx.x & 31;
  int sel  = wid >> 13;            // 0 = user, 1 = pos item, 2 = neg item
  int b    = wid & (BATCH_C - 1);

  float* out_u = out + 2 * BATCH_C;                 // after pos/neg scores
  const float* srcrow;
  float* accrow;
  float* outrow;
  if (sel == 0) {
    srcrow = user_emb + (size_t)uidx[b] * EMB;
    accrow = ue + (size_t)b * EMB;
    outrow = out_u + (size_t)b * EMB;
  } else if (sel == 1) {
    srcrow = item_emb + (size_t)pidx[b] * EMB;
    accrow = pe + (size_t)b * EMB;
    outrow = out_u + (size_t)(BATCH_C + b) * EMB;
  } else {
    srcrow = item_emb + (size_t)nidx[b] * EMB;
    accrow = ne + (size_t)b * EMB;
    outrow = out_u + (size_t)(2 * BATCH_C + b) * EMB;
  }
  const float4* s = (const float4*)srcrow;
  float4 a0 = s[lane];
  float4 a1 = s[lane + 32];
  ((float4*)accrow)[lane]      = a0;
  ((float4*)accrow)[lane + 32] = a1;
  ((float4*)outrow)[lane]      = a0;
  ((float4*)outrow)[lane + 32] = a1;
}

// ---------------------------------------------------------------------------
// Propagation: one wave32 per DIRECTED edge (2M waves). Each lane loads two
// float4 (8 floats) from the source row and does 8 hw fp32 atomic adds into
// the destination row, scaled by dinv[src]*dinv[dst].
// ---------------------------------------------------------------------------
__global__ void lgcn_scatter_kernel(const float* __restrict__ x,
                                    float* __restrict__ y,
                                    const int* __restrict__ erow,
                                    const int* __restrict__ ecol,
                                    const float* __restrict__ dinv) {
  int wid  = (blockIdx.x * blockDim.x + threadIdx.x) >> 5;  // 0..2*N_EDGES-1
  int lane = threadIdx.x & 31;
  bool fwd = wid < N_EDGES_C;
  int e = fwd ? wid : wid - N_EDGES_C;
  int u = erow[e];
  int v = N_USERS_C + ecol[e];
  int s = fwd ? u : v;
  int d = fwd ? v : u;
  float w = dinv[s] * dinv[d];

  const float4* sp = (const float4*)(x + (size_t)s * EMB);
  float4 a = sp[lane];
  float4 b = sp[lane + 32];
  float* dp = y + (size_t)d * EMB + lane * 4;
  unsafeAtomicAdd(dp + 0, w * a.x);
  unsafeAtomicAdd(dp + 1, w * a.y);
  unsafeAtomicAdd(dp + 2, w * a.z);
  unsafeAtomicAdd(dp + 3, w * a.w);
  unsafeAtomicAdd(dp + 128 + 0, w * b.x);
  unsafeAtomicAdd(dp + 128 + 1, w * b.y);
  unsafeAtomicAdd(dp + 128 + 2, w * b.z);
  unsafeAtomicAdd(dp + 128 + 3, w * b.w);
}

// ---------------------------------------------------------------------------
// Per-layer gathered accumulation: acc[b] += x_layer[node_idx[b]]. One wave
// per (selector, batch) row; rows are unique so plain read-modify-write.
// ---------------------------------------------------------------------------
__global__ void lgcn_gather_acc(const float* __restrict__ x,
                                const int* __restrict__ uidx,
                                const int* __restrict__ pidx,
                                const int* __restrict__ nidx,
                                float* __restrict__ ue,
                                float* __restrict__ pe,
                                float* __restrict__ ne) {
  int wid  = (blockIdx.x * blockDim.x + threadIdx.x) >> 5;
  int lane = threadIdx.x & 31;
  int sel  = wid >> 13;
  int b    = wid & (BATCH_C - 1);

  int node;
  float* accrow;
  if (sel == 0)      { node = uidx[b];             accrow = ue + (size_t)b * EMB; }
  else if (sel == 1) { node = N_USERS_C + pidx[b]; accrow = pe + (size_t)b * EMB; }
  else               { node = N_USERS_C + nidx[b]; accrow = ne + (size_t)b * EMB; }

  const float4* s = (const float4*)(x + (size_t)node * EMB);
  float4* a = (float4*)accrow;
  float4 v0 = s[lane], v1 = s[lane + 32];
  float4 c0 = a[lane], c1 = a[lane + 32];
  c0.x += v0.x; c0.y += v0.y; c0.z += v0.z; c0.w += v0.w;
  c1.x += v1.x; c1.y += v1.y; c1.z += v1.z; c1.w += v1.w;
  a[lane]      = c0;
  a[lane + 32] = c1;
}

// ---------------------------------------------------------------------------
// Scores via V_WMMA_F32_16X16X4_F32: each wave computes 16 dot products.
// A (16xK) rows = user accumulators, B (Kx16) cols = item accumulators.
// ISA VGPR layouts: lane l supplies, for both A and B, the float2 at column
// k0 + 2*(l/16) of row base + (l%16). Diagonal of D = the 16 dots.
// Grid launched so EXEC is all-ones at every WMMA (wave32 requirement).
// ---------------------------------------------------------------------------
__global__ void lgcn_scores_wmma(const float* __restrict__ ue,
                                 const float* __restrict__ pe,
                                 const float* __restrict__ ne,
                                 float* __restrict__ out) {
  int wid  = (blockIdx.x * blockDim.x + threadIdx.x) >> 5;  // 0..1023
  int lane = threadIdx.x & 31;
  bool is_neg = wid >= (BATCH_C / 16);
  int base = (is_neg ? wid - BATCH_C / 16 : wid) * 16;

  int row  = base + (lane & 15);
  int koff = (lane >> 4) * 2;   // 0 or 2
  const float* arow = ue + (size_t)row * EMB;
  const float* brow = (is_neg ? ne : pe) + (size_t)row * EMB;

  v8f c = {0.f, 0.f, 0.f, 0.f, 0.f, 0.f, 0.f, 0.f};
#pragma unroll 4
  for (int k0 = 0; k0 < EMB; k0 += 4) {
    v2f av = *(const v2f*)(arow + k0 + koff);
    v2f bv = *(const v2f*)(brow + k0 + koff);
    c = __builtin_amdgcn_wmma_f32_16x16x4_f32(
        /*neg_a=*/false, av, /*neg_b=*/false, bv,
        /*c_mod=*/(short)0, c, /*reuse_a=*/false, /*reuse_b=*/false);
  }

  // Diagonal (m,m): held by lane m (m<8) in c[m], or lane m+16 (m>=8) in c[m-8].
  float diag = 0.f;
#pragma unroll
  for (int v = 0; v < 8; ++v) {
    if (lane == v || lane == v + 24) diag = c[v];
  }
  float* so = out + (is_neg ? BATCH_C : 0);
  if (lane < 8)        so[base + lane]      = diag * 0.0625f;  // acc/4 . acc/4
  else if (lane >= 24) so[base + lane - 16] = diag * 0.0625f;
}

// ---------------------------------------------------------------------------
extern "C" void kernel_launch(void* const* d_in, const int* in_sizes, int n_in,
                              void* d_out, int out_size, void* d_ws, size_t ws_size,
                              hipStream_t stream) {
  const float* user_emb = (const float*)d_in[0];
  const float* item_emb = (const float*)d_in[1];
  const int*   erow     = (const int*)d_in[2];
  const int*   ecol     = (const int*)d_in[3];
  const int*   uidx     = (const int*)d_in[4];
  const int*   pidx     = (const int*)d_in[5];
  const int*   nidx     = (const int*)d_in[6];
  float* out = (float*)d_out;

  // Workspace layout (floats): deg | x0 | x1 | ue | pe | ne   (~641 MB)
  float* ws = (float*)d_ws;
  size_t off = 0;
  float* deg = ws + off; off += (size_t)NUM_NODES;
  float* x0  = ws + off; off += (size_t)NUM_NODES * EMB;
  float* x1  = ws + off; off += (size_t)NUM_NODES * EMB;
  float* ue  = ws + off; off += (size_t)BATCH_C * EMB;
  float* pe  = ws + off; off += (size_t)BATCH_C * EMB;
  float* ne  = ws + off; off += (size_t)BATCH_C * EMB;

  // 1) degrees -> dinv
  hipMemsetAsync(deg, 0, (size_t)NUM_NODES * sizeof(float), stream);
  lgcn_deg_kernel<<<(N_EDGES_C + 255) / 256, 256, 0, stream>>>(erow, ecol, deg);
  lgcn_dinv_kernel<<<(NUM_NODES + 255) / 256, 256, 0, stream>>>(deg);

  // 2) x0 = concat(user_emb, item_emb)
  hipMemcpyAsync(x0, user_emb, (size_t)N_USERS_C * EMB * sizeof(float),
                 hipMemcpyDeviceToDevice, stream);
  hipMemcpyAsync(x0 + (size_t)N_USERS_C * EMB, item_emb,
                 (size_t)N_ITEMS_C * EMB * sizeof(float),
                 hipMemcpyDeviceToDevice, stream);

  // 3) layer-0 accumulators + raw-gather outputs
  lgcn_init_gather<<<(3 * BATCH_C * 32) / 256, 256, 0, stream>>>(
      user_emb, item_emb, uidx, pidx, nidx, ue, pe, ne, out);

  // 4) three propagation layers with per-layer gathered accumulation
  float* xa = x0;
  float* xb = x1;
  for (int layer = 0; layer < 3; ++layer) {
    hipMemsetAsync(xb, 0, (size_t)NUM_NODES * EMB * sizeof(float), stream);
    lgcn_scatter_kernel<<<(2 * N_EDGES_C * 32) / 256, 256, 0, stream>>>(
        xa, xb, erow, ecol, deg);
    lgcn_gather_acc<<<(3 * BATCH_C * 32) / 256, 256, 0, stream>>>(
        xb, uidx, pidx, nidx, ue, pe, ne);
    float* t = xa; xa = xb; xb = t;
  }

  // 5) WMMA dot-product scores (includes the /(N_LAYERS+1)^2 = 1/16 factor)
  lgcn_scores_wmma<<<(2 * (BATCH_C / 16) * 32) / 256, 256, 0, stream>>>(
      ue, pe, ne, out);
}